// CaTentLayer_2594160247024
// MI455X (gfx1250) — compile-verified
//
#include <hip/hip_runtime.h>

typedef __attribute__((ext_vector_type(2))) float v2f;
typedef __attribute__((ext_vector_type(8))) float v8f;

#define T_DIM  16384
#define NC_DIM 2048
#define FW_DIM 100
#define EPS_V  1e-12f

#define CH     16            // channels per workgroup (1 per wave)
#define TILES  2             // 256-output WMMA tiles per wave
#define TBLK   (256 * TILES) // 512 outputs per workgroup
#define WIN    (TBLK + FW_DIM)   // 612-element x window per channel (incl. padded K col)
#define NQ     29            // K = 116 = 29 * 4
#define XROW   653           // padded LDS row: max pcol = 611 + 38 = 649, odd stride
#define WROW   101
#define OROW   257

__global__ __launch_bounds__(512) void catent_fir_wmma(
    const float* __restrict__ x, const float* __restrict__ weights,
    const float* __restrict__ biases, float* __restrict__ out)
{
    __shared__ float xs[CH * XROW];  // x windows, pad-swizzled: pcol = col + (col>>4)
    __shared__ float ws[CH * WROW];  // normalized filters
    __shared__ float os[CH * OROW];  // output staging per tile

    const int tid    = threadIdx.x;
    const int lane   = tid & 31;
    const int wave   = tid >> 5;                 // channel within workgroup
    const int cbase  = blockIdx.y * CH;
    const int tblock = blockIdx.x * TBLK;

    // ---- Phase A1: cooperative coalesced load of x window -> LDS (zero-padded at edges)
    {
        const int ci    = tid & 15;              // channel (fastest -> coalesced)
        const int tstep = tid >> 4;              // 0..31
        const int c     = cbase + ci;
        const int tw0   = tblock - (FW_DIM - 1);
        // pad-swizzled column: (base+tstep) + ((base+tstep)>>4)
        //   = [base + base/16] + [tstep + (tstep>>4)]   (base multiple of 32)
        const int tp    = tstep + (tstep >> 4);
        const int lrow  = ci * XROW;
        #pragma unroll
        for (int base = 0; base < WIN; base += 32) {
            int idx = base + tstep;
            if (idx < WIN) {
                int t = tw0 + idx;
                float v = 0.0f;
                if (t >= 0 && t < T_DIM) v = x[t * NC_DIM + c];
                xs[lrow + (base + (base >> 4)) + tp] = v;
            }
        }
    }

    // ---- Phase A2: per-wave relu + L2-normalize its channel's filter -> LDS
    {
        const int c = cbase + wave;
        float wp[4];
        float ss = 0.0f;
        #pragma unroll
        for (int i = 0; i < 4; ++i) {
            int k = lane + 32 * i;
            float v = 0.0f;
            if (k < FW_DIM) v = fmaxf(weights[k * NC_DIM + c], 0.0f);
            wp[i] = v;
            ss += v * v;
        }
        #pragma unroll
        for (int off = 16; off > 0; off >>= 1) ss += __shfl_xor(ss, off, 32);
        float scale = rsqrtf(fmaxf(ss, EPS_V));
        #pragma unroll
        for (int i = 0; i < 4; ++i) {
            int k = lane + 32 * i;
            if (k < FW_DIM) ws[wave * WROW + k] = wp[i] * scale;
        }
    }
    __syncthreads();

    // ---- Phase B: Toeplitz B fragments of the filter, held in registers.
    // Lane layout (16x4 f32 WMMA operands): n = lane&15, koff = (lane>>4)*2.
    const int n    = lane & 15;
    const int koff = (lane >> 4) * 2;
    v2f bfrag[NQ];
    #pragma unroll
    for (int q = 0; q < NQ; ++q) {
        int j0 = 4 * q + koff;
        int i0 = j0 - n, i1 = j0 + 1 - n;
        float b0 = (i0 >= 0 && i0 < FW_DIM) ? ws[wave * WROW + i0] : 0.0f;
        float b1 = (i1 >= 0 && i1 < FW_DIM) ? ws[wave * WROW + i1] : 0.0f;
        v2f b; b.x = b0; b.y = b1;
        bfrag[q] = b;
    }

    const int m = lane & 15;   // A-matrix row

    for (int tile = 0; tile < TILES; ++tile) {
        v8f acc0 = {};
        v8f acc1 = {};
        // A-fragment base: pcol(colbase + j) = colbase + colbase/16 + j + (j>>4).
        // colbase = tile*256 + 16m (multiple of 16); j = 4q+koff never straddles
        // a 16-boundary within a (j, j+1) pair, and (j>>4) == (q>>2) for koff in {0,2}.
        const int colbase = tile * 256 + 16 * m;
        const int pbase   = wave * XROW + colbase + (colbase >> 4) + koff;
        #pragma unroll
        for (int q = 0; q < NQ; ++q) {
            const int off = 4 * q + (q >> 2);   // compile-time constant per q
            v2f a;
            a.x = xs[pbase + off];
            a.y = xs[pbase + off + 1];
            if (q & 1)
                acc1 = __builtin_amdgcn_wmma_f32_16x16x4_f32(
                    false, a, false, bfrag[q], (short)0, acc1, false, false);
            else
                acc0 = __builtin_amdgcn_wmma_f32_16x16x4_f32(
                    false, a, false, bfrag[q], (short)0, acc0, false, false);
        }
        v8f acc = acc0 + acc1;

        __syncthreads();   // previous tile's staging fully consumed
        // D layout: VGPR v, lanes 0-15 -> row v, lanes 16-31 -> row v+8; col = lane&15
        const int obase = wave * OROW + (lane >> 4) * 128 + n;
        #pragma unroll
        for (int v = 0; v < 8; ++v)
            os[obase + 16 * v] = acc[v];
        __syncthreads();

        // ---- coalesced global store with fused bias
        {
            const int ci    = tid & 15;
            const int tstep = tid >> 4;
            const float b   = biases[cbase + ci];
            const int lbase = ci * OROW + tstep;
            const int gbase = (tblock + tile * 256 + tstep) * NC_DIM + cbase + ci;
            #pragma unroll
            for (int p = 0; p < 256; p += 32)
                out[gbase + p * NC_DIM] = os[lbase + p] + b;
        }
    }
}

extern "C" void kernel_launch(void* const* d_in, const int* in_sizes, int n_in,
                              void* d_out, int out_size, void* d_ws, size_t ws_size,
                              hipStream_t stream) {
    (void)in_sizes; (void)n_in; (void)out_size; (void)d_ws; (void)ws_size;
    const float* x  = (const float*)d_in[0];
    const float* w  = (const float*)d_in[1];
    const float* b  = (const float*)d_in[2];
    float* out      = (float*)d_out;
    dim3 grid(T_DIM / TBLK, NC_DIM / CH);   // 32 x 128
    catent_fir_wmma<<<grid, 512, 0, stream>>>(x, w, b, out);
}